// DGLossVer2_42992622633087
// MI455X (gfx1250) — compile-verified
//
#include <hip/hip_runtime.h>
#include <hip/hip_bf16.h>
#include <stdint.h>

// DGLossVer2 for MI455X (gfx1250, wave32).
// Memory-bound f32 op (~135 MB / 23.3 TB/s ~ 6 us floor). Fused heterogeneous
// kernel: gyro blocks (VALU-heavy, 25 MB) co-resident with gnll blocks
// (BW-heavy, 100 MB) so compute hides under streaming. CDNA5 paths:
//  - global_load_async_to_lds_b128 + s_wait_asynccnt (coalesced staging of the
//    per-thread 384B rotation windows into padded LDS rows)
//  - v_wmma_f32_16x16x4_f32 as an exact lane^16 fold in the final reduction
//  - global_prefetch via __builtin_prefetch in the streaming NLL role

constexpr int   kT        = 16384;
constexpr float kDT       = 0.005f;
constexpr float kHuberInv = 200.0f;  // 1/HUBER
constexpr float kWH2      = 25.0f;   // W * HUBER^2 = 1e6 * 0.005^2

constexpr int kGyroBlocks = 512;     // 128 seq * 4 quarters
constexpr int kNllBlocks  = 1024;
constexpr int kThreads    = 128;

typedef __attribute__((ext_vector_type(2))) float v2f;
typedef __attribute__((ext_vector_type(8))) float v8f;

__device__ __forceinline__ void so3_exp3(float x, float y, float z, float* R) {
  float a2 = x * x + y * y + z * z;
  bool small = a2 < 1e-12f;
  float a2s = small ? 1.0f : a2;
  float a = __builtin_sqrtf(a2s);
  float sa, ca;
  __sincosf(a, &sa, &ca);
  float st = small ? (1.0f - a2 * (1.0f / 6.0f)) : (sa / a);
  float ct = small ? (0.5f - a2 * (1.0f / 24.0f)) : ((1.0f - ca) / a2s);
  float xx = x * x, yy = y * y, zz = z * z;
  float xy = x * y, xz = x * z, yz = y * z;
  R[0] = 1.0f - ct * (yy + zz); R[1] = ct * xy - st * z; R[2] = ct * xz + st * y;
  R[3] = ct * xy + st * z; R[4] = 1.0f - ct * (xx + zz); R[5] = ct * yz - st * x;
  R[6] = ct * xz - st * y; R[7] = ct * yz + st * x; R[8] = 1.0f - ct * (xx + yy);
}

__device__ __forceinline__ void mm3(const float* A, const float* B, float* C) {
#pragma unroll
  for (int r = 0; r < 3; ++r)
#pragma unroll
    for (int c = 0; c < 3; ++c)
      C[3 * r + c] = A[3 * r + 0] * B[c] + A[3 * r + 1] * B[3 + c] + A[3 * r + 2] * B[6 + c];
}

__device__ __forceinline__ float smooth_l1(float u) {
  float au = fabsf(u);
  return (au < 1.0f) ? (0.5f * au * au) : (au - 0.5f);
}

// sum_c smooth_l1( so3_log(A^T B)[c] / HUBER )
__device__ __forceinline__ float huber_of_log_mtm(const float* A, const float* B) {
  float M[9];
#pragma unroll
  for (int r = 0; r < 3; ++r)
#pragma unroll
    for (int c = 0; c < 3; ++c)
      M[3 * r + c] = A[r] * B[c] + A[3 + r] * B[3 + c] + A[6 + r] * B[6 + c];
  float tr = M[0] + M[4] + M[8];
  float cc = 0.5f * (tr - 1.0f);
  cc = fminf(fmaxf(cc, -1.0f + 1e-6f), 1.0f - 1e-6f);
  float ang = acosf(cc);
  float f = (ang < 1e-6f) ? 0.5f : (ang / (2.0f * __sinf(ang)));
  float w0 = f * (M[7] - M[5]);
  float w1 = f * (M[2] - M[6]);
  float w2 = f * (M[3] - M[1]);
  return smooth_l1(w0 * kHuberInv) + smooth_l1(w1 * kHuberInv) + smooth_l1(w2 * kHuberInv);
}

__device__ __forceinline__ float warp_sum32(float v) {
#pragma unroll
  for (int m = 16; m >= 1; m >>= 1) v += __shfl_xor(v, m, 32);
  return v;
}

__device__ __forceinline__ float gterm(float d, float s) {
  float v = s * s;
  v = (v < 1e-6f) ? 1e-6f : v;
  return __logf(v) + d * d / v;
}

// ---------------------------------------------------------------------------
// Fused kernel: blocks [0,512) = gyro role (1/4 sequence each, async-staged
// through LDS); blocks [512,1536) = gnll streaming role. Both roles coexist
// on WGPs so gyro VALU work hides under gnll HBM streaming.
// ---------------------------------------------------------------------------
__global__ __launch_bounds__(kThreads) void dg_fused_kernel(
    const float* __restrict__ w_hat, const float* __restrict__ dw16,
    const float* __restrict__ w_gt, const float* __restrict__ w_mean,
    const float* __restrict__ w_std, float* __restrict__ ws) {
  __shared__ float lds[kThreads * 100];  // 96 data floats + 4 pad per thread row
  __shared__ float red[8];

  const int tid = threadIdx.x;
  const int blk = blockIdx.x;

  if (blk < kGyroBlocks) {
    // ================= gyro role =================
    const int seq = blk >> 2;
    const int q   = blk & 3;

    // --- async coalesced stage: 3072 float4 tiles into padded LDS rows ---
    const float4* gsrc = reinterpret_cast<const float4*>(
        w_hat + ((size_t)seq * kT + (size_t)q * 4096) * 3);
    unsigned lbase = (unsigned)(uintptr_t)(&lds[0]);  // low 32 bits = LDS offset
#pragma unroll
    for (int k = 0; k < 24; ++k) {
      int i = tid + k * kThreads;       // float4 index in [0, 3072)
      int c = i / 24;                   // destination thread-row
      int w = i - c * 24;               // float4 within row
      unsigned loff = lbase + (unsigned)(c * 100 + w * 4) * 4u;
      const float4* gp = gsrc + i;
      asm volatile("global_load_async_to_lds_b128 %0, %1, off"
                   :: "v"(loff), "v"(gp) : "memory");
    }
    asm volatile("s_wait_asynccnt 0" ::: "memory");
    __syncthreads();

    // --- sequential products of 16 exp(DT*w) (== the pairwise tree) ---
    const float* my = &lds[tid * 100];
    float Pa[9], Pb[9], R[9], Tm[9];
    so3_exp3(kDT * my[0], kDT * my[1], kDT * my[2], Pa);
    for (int i = 1; i < 16; ++i) {
      so3_exp3(kDT * my[3 * i], kDT * my[3 * i + 1], kDT * my[3 * i + 2], R);
      mm3(Pa, R, Tm);
#pragma unroll
      for (int j = 0; j < 9; ++j) Pa[j] = Tm[j];
    }
    so3_exp3(kDT * my[48], kDT * my[49], kDT * my[50], Pb);
    for (int i = 17; i < 32; ++i) {
      so3_exp3(kDT * my[3 * i], kDT * my[3 * i + 1], kDT * my[3 * i + 2], R);
      mm3(Pb, R, Tm);
#pragma unroll
      for (int j = 0; j < 9; ++j) Pb[j] = Tm[j];
    }

    // --- ground-truth rotations at the 16-subsampled points ---
    const int j2 = q * kThreads + tid;  // in-sequence index in [0,512)
    const float* dwp = dw16 + ((size_t)seq * kT + (size_t)j2 * 32) * 3;
    float Ga[9], Gb[9];
    so3_exp3(dwp[0], dwp[1], dwp[2], Ga);
    so3_exp3(dwp[48], dwp[49], dwp[50], Gb);   // +16 samples -> +48 floats

    float s16 = 0.0f, s32 = 0.0f;
    if (2 * j2     >= 5) s16 += huber_of_log_mtm(Pa, Ga);  // j = 2*j2
    if (2 * j2 + 1 >= 5) s16 += huber_of_log_mtm(Pb, Gb);  // j = 2*j2+1
    float P32[9], G32[9];
    mm3(Pa, Pb, P32);
    mm3(Ga, Gb, G32);
    if (j2 >= 5) s32 = huber_of_log_mtm(P32, G32);

    // --- block reduction, deterministic partials to d_ws ---
    s16 = warp_sum32(s16);
    s32 = warp_sum32(s32);
    if ((tid & 31) == 0) { red[tid >> 5] = s16; red[4 + (tid >> 5)] = s32; }
    __syncthreads();
    if (tid == 0) {
      ws[blk]               = red[0] + red[1] + red[2] + red[3];
      ws[kGyroBlocks + blk] = red[4] + red[5] + red[6] + red[7];
    }
  } else {
    // ================= gnll role =================
    const float4* h4 = (const float4*)w_hat;
    const float4* g4 = (const float4*)w_gt;
    const float4* m4 = (const float4*)w_mean;
    const float4* s4 = (const float4*)w_std;
    const int b = blk - kGyroBlocks;               // [0,1024)
    const int stride = kNllBlocks * kThreads;      // 131072 float4
    const int gid = b * kThreads + tid;
    float acc = 0.0f;
#pragma unroll 2
    for (int k = 0; k < 12; ++k) {                 // 12 * 131072 = 1,572,864
      int i = gid + k * stride;
      if (k + 2 < 12) {
        __builtin_prefetch(h4 + i + 2 * stride, 0, 1);
        __builtin_prefetch(g4 + i + 2 * stride, 0, 1);
        __builtin_prefetch(m4 + i + 2 * stride, 0, 1);
        __builtin_prefetch(s4 + i + 2 * stride, 0, 1);
      }
      float4 h = h4[i], g = g4[i], m = m4[i], s = s4[i];
      acc += gterm(g.x - h.x - m.x, s.x);
      acc += gterm(g.y - h.y - m.y, s.y);
      acc += gterm(g.z - h.z - m.z, s.z);
      acc += gterm(g.w - h.w - m.w, s.w);
    }
    acc = warp_sum32(acc);
    if ((tid & 31) == 0) red[tid >> 5] = acc;
    __syncthreads();
    if (tid == 0) {
      ws[2 * kGyroBlocks + b] = red[0] + red[1] + red[2] + red[3];
    }
  }
}

// ---------------------------------------------------------------------------
// Final reduce: one wave. WMMA f32 16x16x4 with A=ones, B=[x;0] gives
// D[m][n] = x[n] + x[n+16] (exact lane^16 fold), then 4 xor-shuffles.
// ---------------------------------------------------------------------------
__device__ __forceinline__ float wave32_total(float x) {
#if __has_builtin(__builtin_amdgcn_wmma_f32_16x16x4_f32)
  v2f a; a[0] = 1.0f; a[1] = 1.0f;       // A = ones(16x4)
  v2f b; b[0] = x;    b[1] = 0.0f;       // B rows: K0=x[0..15], K1=x[16..31]
  v8f c = {};
  c = __builtin_amdgcn_wmma_f32_16x16x4_f32(false, a, false, b, (short)0, c,
                                            false, false);
  float s = c[0];                         // lane L holds x[L&15] + x[(L&15)+16]
  s += __shfl_xor(s, 8, 32);
  s += __shfl_xor(s, 4, 32);
  s += __shfl_xor(s, 2, 32);
  s += __shfl_xor(s, 1, 32);
  return s;
#else
  return warp_sum32(x);
#endif
}

__global__ __launch_bounds__(32) void dg_final_kernel(const float* __restrict__ part,
                                                      float* __restrict__ out) {
  const int l = threadIdx.x;
  float a16 = 0.0f, a32 = 0.0f, ag = 0.0f;
  for (int i = l; i < kGyroBlocks; i += 32) {
    a16 += part[i];
    a32 += part[kGyroBlocks + i];
  }
  for (int i = l; i < kNllBlocks; i += 32) {
    ag += part[2 * kGyroBlocks + i];
  }
  float S16 = wave32_total(a16);
  float S32 = wave32_total(a32);
  float Sg  = wave32_total(ag);
  if (l == 0) {
    float gyro16 = kWH2 * S16 / (128.0f * 1019.0f * 3.0f);
    float gyro32 = kWH2 * 0.25f * S32 / (128.0f * 507.0f * 3.0f);
    float gnll   = Sg / (2.0f * 6291456.0f);
    out[0] = gyro16 + gyro32 + gnll;
  }
}

extern "C" void kernel_launch(void* const* d_in, const int* in_sizes, int n_in,
                              void* d_out, int out_size, void* d_ws, size_t ws_size,
                              hipStream_t stream) {
  (void)in_sizes; (void)n_in; (void)out_size; (void)ws_size;
  const float* w_hat  = (const float*)d_in[0];
  const float* dw16   = (const float*)d_in[1];
  const float* w_gt   = (const float*)d_in[2];
  const float* w_mean = (const float*)d_in[3];
  const float* w_std  = (const float*)d_in[4];
  float* ws  = (float*)d_ws;  // [0,512) s16 | [512,1024) s32 | [1024,2048) gnll
  float* out = (float*)d_out;

  dg_fused_kernel<<<kGyroBlocks + kNllBlocks, kThreads, 0, stream>>>(
      w_hat, dw16, w_gt, w_mean, w_std, ws);
  dg_final_kernel<<<1, 32, 0, stream>>>(ws, out);
}